// GraphConv_34660386078858
// MI455X (gfx1250) — compile-verified
//
#include <hip/hip_runtime.h>
#include <hip/hip_bf16.h>

typedef __attribute__((ext_vector_type(2))) float v2f;
typedef __attribute__((ext_vector_type(8))) float v8f;

#define DIM 128

// ---------------------------------------------------------------------------
// Kernel 1: zero the aggregation workspace (agg = 0), float4 wide.
// ---------------------------------------------------------------------------
__global__ void gc_zero_kernel(float4* __restrict__ p, int n4) {
    int i = blockIdx.x * blockDim.x + threadIdx.x;
    if (i < n4) {
        float4 z; z.x = 0.f; z.y = 0.f; z.z = 0.f; z.w = 0.f;
        p[i] = z;
    }
}

// ---------------------------------------------------------------------------
// Kernel 2: edge scatter.  One wave32 per edge: lane l handles float4 chunk l
// (32 chunks * 4 floats = 128).  Loads are fully coalesced 512B rows; the
// scatter uses hardware fire-and-forget global_atomic_add_f32 (no return).
// x and agg both fit in L2 (192MB), so this is L2-atomic-throughput bound.
// ---------------------------------------------------------------------------
__global__ void gc_scatter_kernel(const float* __restrict__ x,
                                  const int*   __restrict__ esrc,
                                  const int*   __restrict__ edst,
                                  const float* __restrict__ ew,
                                  float*       __restrict__ agg,
                                  int n_edges) {
    int edge = (blockIdx.x * blockDim.x + threadIdx.x) >> 5;
    int lane = threadIdx.x & 31;
    if (edge >= n_edges) return;

    int   s = esrc[edge];          // uniform per wave -> scalar load
    int   d = edst[edge];
    float w = ew[edge];

    const float4* xr = (const float4*)(x + (size_t)s * DIM);
    float4 v = xr[lane];           // global_load_b128, coalesced

    float* ar = agg + (size_t)d * DIM + lane * 4;
    unsafeAtomicAdd(ar + 0, v.x * w);   // global_atomic_add_f32 (no return)
    unsafeAtomicAdd(ar + 1, v.y * w);
    unsafeAtomicAdd(ar + 2, v.z * w);
    unsafeAtomicAdd(ar + 3, v.w * w);
}

// ---------------------------------------------------------------------------
// Kernel 3: out = relu(agg @ W^T) via V_WMMA_F32_16X16X4_F32 (full f32).
//
// Grid: one block per 16-row M tile (625 blocks).  Block = 256 threads =
// 8 waves; wave wv owns the 16x16 output tile at columns [wv*16, wv*16+16).
// K loop: 128/4 = 32 WMMA ops per wave.
//
// f32 WMMA VGPR layouts (ISA 7.12.2):
//   A 16x4 : lanes 0-15 -> rows M=lane, regs {K, K+1}; lanes 16-31 -> {K+2, K+3}
//   B 4x16 : lanes 0-15 -> cols N=lane, regs {K, K+1}; lanes 16-31 -> {K+2, K+3}
//   C/D    : reg v -> row v (lanes 0-15) / row v+8 (lanes 16-31), col = lane%16
//
// B[k][n] = W[n][k] (W is [out_dim, in_dim] row-major), so each lane streams
// a contiguous float2 from row n of W -- W (64KB) stays L0/L2 resident.
// ---------------------------------------------------------------------------
__global__ void gc_gemm_relu_kernel(const float* __restrict__ agg, // [M,128]
                                    const float* __restrict__ W,   // [128,128]
                                    float*       __restrict__ out, // [M,128]
                                    int M) {
    const int lane  = threadIdx.x & 31;
    const int wv    = threadIdx.x >> 5;       // 0..7 : N tile index
    const int half  = lane >> 4;              // 0 or 1 (K sub-pair select)
    const int l     = lane & 15;
    const int mbase = blockIdx.x * 16;
    const int nbase = wv * 16;

    int mload = mbase + l;
    if (mload > M - 1) mload = M - 1;         // keep EXEC all-1s for WMMA

    const float* arow = agg + (size_t)mload * DIM + half * 2;
    const float* brow = W   + (size_t)(nbase + l) * DIM + half * 2;

    v8f c = {};
#pragma unroll
    for (int k = 0; k < DIM; k += 4) {
        v2f a, b;
        a.x = arow[k];     // K = k + 2*half
        a.y = arow[k + 1]; // K = k + 2*half + 1
        b.x = brow[k];
        b.y = brow[k + 1];
        c = __builtin_amdgcn_wmma_f32_16x16x4_f32(
                /*neg_a=*/false, a, /*neg_b=*/false, b,
                /*c_mod=*/(short)0, c, /*reuse_a=*/false, /*reuse_b=*/false);
    }

    const int n = nbase + l;
#pragma unroll
    for (int v = 0; v < 8; ++v) {
        int m = mbase + v + half * 8;
        if (m < M) {
            float val = c[v];
            out[(size_t)m * DIM + n] = val > 0.f ? val : 0.f;
        }
    }
}

// ---------------------------------------------------------------------------
// Launch: inputs are {x, edge_src, edge_dst, edge_weight, W}; d_ws holds agg.
// ---------------------------------------------------------------------------
extern "C" void kernel_launch(void* const* d_in, const int* in_sizes, int n_in,
                              void* d_out, int out_size, void* d_ws, size_t ws_size,
                              hipStream_t stream) {
    const float* x    = (const float*)d_in[0];
    const int*   esrc = (const int*)  d_in[1];
    const int*   edst = (const int*)  d_in[2];
    const float* ew   = (const float*)d_in[3];
    const float* W    = (const float*)d_in[4];
    float*       out  = (float*)d_out;
    float*       agg  = (float*)d_ws;        // [N, 128] f32 scratch

    const int N = in_sizes[0] / DIM;         // 10000 nodes
    const int E = in_sizes[1];               // 640000 edges

    // 1) agg = 0
    int n4 = N * (DIM / 4);
    gc_zero_kernel<<<(n4 + 255) / 256, 256, 0, stream>>>((float4*)agg, n4);

    // 2) agg[dst] += w * x[src]   (one wave per edge, 8 edges per block)
    gc_scatter_kernel<<<(E + 7) / 8, 256, 0, stream>>>(x, esrc, edst, ew, agg, E);

    // 3) out = relu(agg @ W^T)    (one 16-row M tile per block, 8 N tiles/waves)
    gc_gemm_relu_kernel<<<(N + 15) / 16, 256, 0, stream>>>(agg, W, out, N);
}